// QuantizationLayer_17437567222232
// MI455X (gfx1250) — compile-verified
//
#include <hip/hip_runtime.h>

// ---------------------------------------------------------------------------
// Event-camera quantization layer for MI455X (gfx1250, wave32).
// Pipeline: memset -> tmax -> scatter -> slice0 -> 15x erode(WMMA) -> argmax
//           -> assemble.
// ---------------------------------------------------------------------------

namespace {
constexpr int kH  = 240;
constexpr int kW  = 304;
constexpr int kS  = 16;
constexpr int kB  = 8;
constexpr int kHh = kH / 2;   // 120
constexpr int kWh = kW / 2;   // 152
}

typedef __attribute__((ext_vector_type(2))) float v2f;
typedef __attribute__((ext_vector_type(8))) float v8f;

// ---------------------------------------------------------------------------
// Kernel 1: per-batch max of t (bits trick: t > 0, so int max == float max).
// ---------------------------------------------------------------------------
__global__ void k_tmax(const float* __restrict__ ev, int* __restrict__ tmax_bits, int N) {
    __shared__ int lmax[kB];
    if (threadIdx.x < kB) lmax[threadIdx.x] = 0;
    __syncthreads();
    for (int i = blockIdx.x * blockDim.x + threadIdx.x; i < N;
         i += gridDim.x * blockDim.x) {
        float t = ev[(size_t)i * 5 + 2];
        int   b = (int)ev[(size_t)i * 5 + 4];
        atomicMax(&lmax[b], __float_as_int(t));
    }
    __syncthreads();
    if (threadIdx.x < kB) atomicMax(&tmax_bits[threadIdx.x], lmax[threadIdx.x]);
}

// ---------------------------------------------------------------------------
// Kernel 2: scatter. Dilution = plain store of 1.0 (idempotent); container /
// counter / timer need float atomics (L2-resident, ~14-55 events per cell).
// ---------------------------------------------------------------------------
__global__ void k_scatter(const float* __restrict__ ev, const int* __restrict__ tmax_bits,
                          float* __restrict__ dil, float* __restrict__ cont,
                          float* __restrict__ cnt_h, float* __restrict__ tim_h, int N) {
    int i = blockIdx.x * blockDim.x + threadIdx.x;
    if (i >= N) return;
    // gfx1250 prefetch path (global_prefetch_b8) for the streaming event read.
    __builtin_prefetch(ev + (size_t)i * 5 + 8192, 0, 1);
    float fx = ev[(size_t)i * 5 + 0];
    float fy = ev[(size_t)i * 5 + 1];
    float t  = ev[(size_t)i * 5 + 2];
    float fb = ev[(size_t)i * 5 + 4];
    int x = (int)fx, y = (int)fy, b = (int)fb;
    float tn = t / __int_as_float(tmax_bits[b]);
    int ts = (int)(tn * (float)kS);
    ts = ts > kS - 1 ? kS - 1 : ts;

    dil[(size_t)x + (size_t)kW * y + (size_t)kW * kH * ts + (size_t)kW * kH * kS * b] = 1.0f;
    atomicAdd(cont + (size_t)x + (size_t)kW * y + (size_t)kW * kH * b, 1.0f);
    size_t hidx = (size_t)(x >> 1) + (size_t)kWh * (y >> 1) + (size_t)kWh * kHh * b;
    atomicAdd(cnt_h + hidx, 1.0f);
    atomicAdd(tim_h + hidx, tn);
}

// ---------------------------------------------------------------------------
// Kernel 3: slice 0 -> copy to ping-pong buf 0, zero-count, downsample store.
// ---------------------------------------------------------------------------
__global__ void k_slice0(const float* __restrict__ dil, float* __restrict__ pp0,
                         float* __restrict__ ds, int* __restrict__ zero_cnt) {
    int b = blockIdx.y;
    int p = blockIdx.x * blockDim.x + threadIdx.x;
    int zero = 0;
    if (p < kH * kW) {
        float v = dil[((size_t)b * kS + 0) * kH * kW + p];
        pp0[(size_t)b * kH * kW + p] = v;
        zero = (v <= 0.0f) ? 1 : 0;
        int y = p / kW, x = p % kW;
        if (((x | y) & 1) == 0)
            ds[(((size_t)b * kS + 0) * kHh + (y >> 1)) * kWh + (x >> 1)] = v;
    }
    __shared__ int red[256];
    red[threadIdx.x] = zero;
    __syncthreads();
    for (int s = 128; s; s >>= 1) {
        if (threadIdx.x < s) red[threadIdx.x] += red[threadIdx.x + s];
        __syncthreads();
    }
    if (threadIdx.x == 0) atomicAdd(&zero_cnt[b * kS + 0], red[0]);
}

// ---------------------------------------------------------------------------
// Kernel 4: one erosion step. One wave per 16x16 output tile.
//   v(18x18) = alpha*raw_slice + beta*prev  (zero-padded SAME)
//   R[r][x]  = v[r][x] + v[r][x+1] + v[r][x+2]         (horizontal 3-tap)
//   C        = A(16x20 band) x R(20x16)                (vertical 3-tap, WMMA)
//   out      = (C - v_center) * 0.125 - 0.25
// A-matrix per ISA 16x4 f32 layout: lanes 0-15 hold K=k0,k0+1 of row M=lane;
// lanes 16-31 hold K=k0+2,k0+3 of row M=lane-16. C: 8 VGPRs, M=j (+8 hi half).
// ---------------------------------------------------------------------------
__global__ __launch_bounds__(32)
void k_erode(const float* __restrict__ dil, const float* __restrict__ prev,
             float* __restrict__ cur, float* __restrict__ ds,
             int* __restrict__ zero_cnt, int slice, float alpha, float beta) {
    const int b  = blockIdx.z;
    const int ty = blockIdx.y * 16;
    const int tx = blockIdx.x * 16;
    const int lane = threadIdx.x;

    __shared__ float v[18][20];
    __shared__ float R[20][17];

    const float* raw = dil + ((size_t)b * kS + slice) * kH * kW;
    const float* pv  = prev + (size_t)b * kH * kW;

    // Load 18x18 halo of combined input v (zero pad at image borders).
    for (int idx = lane; idx < 18 * 18; idx += 32) {
        int r = idx / 18, c = idx % 18;
        int gy = ty + r - 1, gx = tx + c - 1;
        float val = 0.0f;
        if (gy >= 0 && gy < kH && gx >= 0 && gx < kW) {
            size_t off = (size_t)gy * kW + gx;
            val = alpha * raw[off] + beta * pv[off];
        }
        v[r][c] = val;
    }
    __syncthreads();

    // Horizontal 3-tap into R; rows 18,19 zero (K padded to 20 for WMMA).
    for (int idx = lane; idx < 20 * 16; idx += 32) {
        int r = idx / 16, x = idx % 16;
        R[r][x] = (r < 18) ? (v[r][x] + v[r][x + 1] + v[r][x + 2]) : 0.0f;
    }
    __syncthreads();

    // Vertical 3-tap as banded matmul: 5 chained V_WMMA_F32_16X16X4_F32.
    const int m     = lane & 15;
    const int khalf = (lane >> 4) << 1;   // 0 for lanes 0-15, 2 for 16-31
    v8f c = {};
#pragma unroll
    for (int kk = 0; kk < 5; ++kk) {
        int k0 = kk * 4 + khalf;
        v2f a, bm;
        a.x  = (k0     >= m && k0     <= m + 2) ? 1.0f : 0.0f;
        a.y  = (k0 + 1 >= m && k0 + 1 <= m + 2) ? 1.0f : 0.0f;
        bm.x = R[k0][m];        // B 4x16: N = lane&15, K rows k0, k0+1
        bm.y = R[k0 + 1][m];
        c = __builtin_amdgcn_wmma_f32_16x16x4_f32(
            /*neg_a=*/false, a, /*neg_b=*/false, bm,
            /*c_mod=*/(short)0, c, /*reuse_a=*/false, /*reuse_b=*/false);
    }

    // Epilogue: subtract center, scale, bias; store + zero-count + downsample.
    float* curb = cur + (size_t)b * kH * kW;
    const int mhi = (lane >> 4) << 3;     // +8 row offset for lanes 16-31
    int myzero = 0;
#pragma unroll
    for (int j = 0; j < 8; ++j) {
        int y = j + mhi;
        int x = m;
        float val = (c[j] - v[y + 1][x + 1]) * 0.125f - 0.25f;
        int gy = ty + y, gx = tx + x;
        curb[(size_t)gy * kW + gx] = val;
        myzero += (val <= 0.0f) ? 1 : 0;
        if (((gy | gx) & 1) == 0)
            ds[(((size_t)b * kS + slice) * kHh + (gy >> 1)) * kWh + (gx >> 1)] = val;
    }
#pragma unroll
    for (int off = 16; off; off >>= 1) myzero += __shfl_down(myzero, off);
    if (lane == 0) atomicAdd(&zero_cnt[b * kS + slice], myzero);
}

// ---------------------------------------------------------------------------
// Kernel 5: per-batch argmax over 16 slices (first-max tie break, as jnp).
// ---------------------------------------------------------------------------
__global__ void k_argmax(const int* __restrict__ zero_cnt, int* __restrict__ best_idx) {
    int b = threadIdx.x;
    if (b < kB) {
        int bi = 0, bv = zero_cnt[b * kS];
        for (int s = 1; s < kS; ++s) {
            int vv = zero_cnt[b * kS + s];
            if (vv > bv) { bv = vv; bi = s; }
        }
        best_idx[b] = bi;
    }
}

// ---------------------------------------------------------------------------
// Kernel 6: assemble (B,5,120,152) output.
// ---------------------------------------------------------------------------
__global__ void k_assemble(const float* __restrict__ cont, const float* __restrict__ cnt_h,
                           const float* __restrict__ tim_h, const float* __restrict__ ds,
                           const int* __restrict__ best_idx, float* __restrict__ out) {
    int idx = blockIdx.x * blockDim.x + threadIdx.x;
    if (idx >= kB * kHh * kWh) return;
    int xh = idx % kWh;
    int yh = (idx / kWh) % kHh;
    int b  = idx / (kWh * kHh);
    const float* cb = cont + (size_t)b * kH * kW;
    int y0 = yh * 2, x0 = xh * 2;
    float c00 = cb[(size_t)y0 * kW + x0];
    float c01 = cb[(size_t)y0 * kW + x0 + 1];
    float c10 = cb[(size_t)(y0 + 1) * kW + x0];
    float c11 = cb[(size_t)(y0 + 1) * kW + x0 + 1];
    float diff_y = (c00 - c10) + (c01 - c11);
    float diff_x = (c10 - c11) + (c00 - c01);
    float cnt = cnt_h[idx];
    float tim = tim_h[idx] / (cnt == 0.0f ? 1.0f : cnt);
    float best = ds[(((size_t)b * kS + best_idx[b]) * kHh + yh) * kWh + xh];
    size_t base = (size_t)b * 5 * kHh * kWh + (size_t)yh * kWh + xh;
    const size_t chs = (size_t)kHh * kWh;
    out[base + 0 * chs] = diff_x;
    out[base + 1 * chs] = diff_y;
    out[base + 2 * chs] = tim;
    out[base + 3 * chs] = cnt;
    out[base + 4 * chs] = best;
}

// ---------------------------------------------------------------------------
extern "C" void kernel_launch(void* const* d_in, const int* in_sizes, int n_in,
                              void* d_out, int out_size, void* d_ws, size_t ws_size,
                              hipStream_t stream) {
    const float* ev = (const float*)d_in[0];
    const int N = in_sizes[0] / 5;
    float* out = (float*)d_out;
    float* ws  = (float*)d_ws;

    // --- workspace layout (element offsets) ---
    const size_t n_dil  = (size_t)kB * kS * kH * kW;   // 9,338,880
    const size_t n_img  = (size_t)kB * kH * kW;        //   583,680
    const size_t n_half = (size_t)kB * kHh * kWh;      //   145,920
    const size_t n_ds   = (size_t)kB * kS * kHh * kWh; // 2,334,720

    size_t o = 0;
    const size_t o_dil  = o; o += n_dil;
    const size_t o_cont = o; o += n_img;
    const size_t o_cnth = o; o += n_half;
    const size_t o_timh = o; o += n_half;
    const size_t o_zc   = o; o += kB * kS;   // ints
    const size_t o_tmax = o; o += kB;        // int bits
    const size_t zero_prefix = o;            // everything above needs zeroing
    const size_t o_pp0  = o; o += n_img;
    const size_t o_pp1  = o; o += n_img;
    const size_t o_ds   = o; o += n_ds;
    const size_t o_best = o; o += kB;        // ints
    if (ws_size < o * sizeof(float)) return;

    float* dil   = ws + o_dil;
    float* cont  = ws + o_cont;
    float* cnth  = ws + o_cnth;
    float* timh  = ws + o_timh;
    int*   zc    = (int*)(ws + o_zc);
    int*   tmaxb = (int*)(ws + o_tmax);
    float* pp0   = ws + o_pp0;
    float* pp1   = ws + o_pp1;
    float* dsbuf = ws + o_ds;
    int*   best  = (int*)(ws + o_best);

    hipMemsetAsync(d_ws, 0, zero_prefix * sizeof(float), stream);

    k_tmax<<<1024, 256, 0, stream>>>(ev, tmaxb, N);
    k_scatter<<<(N + 255) / 256, 256, 0, stream>>>(ev, tmaxb, dil, cont, cnth, timh, N);

    dim3 g0((kH * kW + 255) / 256, kB);
    k_slice0<<<g0, 256, 0, stream>>>(dil, pp0, dsbuf, zc);

    dim3 ge(kW / 16, kH / 16, kB);   // 19 x 15 x 8 tiles, one wave each
    for (int i = 1; i < kS; ++i) {
        float alpha = 0.5f + (float)(kS - i) / (float)kS;
        float beta  = (float)i / (float)kS;
        const float* prevb = (i & 1) ? pp0 : pp1;
        float*       curb  = (i & 1) ? pp1 : pp0;
        k_erode<<<ge, 32, 0, stream>>>(dil, prevb, curb, dsbuf, zc, i, alpha, beta);
    }

    k_argmax<<<1, 32, 0, stream>>>(zc, best);
    k_assemble<<<(kB * kHh * kWh + 255) / 256, 256, 0, stream>>>(cont, cnth, timh,
                                                                 dsbuf, best, out);
}